// LearnableVQ_13271448944640
// MI455X (gfx1250) — compile-verified
//
#include <hip/hip_runtime.h>
#include <hip/hip_bf16.h>

#define B_ 8
#define H_ 8
#define L_ 4096
#define DK_ 64
#define S_ 512

typedef float v2f __attribute__((ext_vector_type(2)));
typedef float v4f __attribute__((ext_vector_type(4)));
typedef float v8f __attribute__((ext_vector_type(8)));

// ---------------------------------------------------------------------------
// Zero the scalar accumulator slots each launch (graph-replay deterministic).
// ---------------------------------------------------------------------------
__global__ void vq_init_scalars(float* lc, float* lcb) {
    *lc  = 0.0f;   // l_commit accumulator
    *lcb = 0.0f;   // l_codebook (numerically zero in the reference)
}

// ---------------------------------------------------------------------------
// Normalize codebook: c = c_sum / max(c_count, 0.01), and c2 = |c|^2.
// One wave per centroid; each lane handles 2 of the 64 elements.
// ---------------------------------------------------------------------------
__global__ __launch_bounds__(256) void vq_prep(const float* __restrict__ c_sum,
                                               const float* __restrict__ c_count,
                                               float* __restrict__ ws_c,
                                               float* __restrict__ ws_c2) {
    const int wid  = threadIdx.x >> 5;
    const int lane = threadIdx.x & 31;
    const int cid  = blockIdx.x * 8 + wid;            // [0, H*S)
    const float inv = 1.0f / fmaxf(c_count[cid], 0.01f);
    v2f cs = *(const v2f*)(c_sum + (size_t)cid * DK_ + 2 * lane);
    v2f cv; cv.x = cs.x * inv; cv.y = cs.y * inv;
    *(v2f*)(ws_c + (size_t)cid * DK_ + 2 * lane) = cv;
    float sq = cv.x * cv.x + cv.y * cv.y;
    #pragma unroll
    for (int off = 16; off > 0; off >>= 1) sq += __shfl_xor(sq, off, 32);
    if (lane == 0) ws_c2[cid] = sq;
}

// ---------------------------------------------------------------------------
// Main kernel: per 16-row tile per wave, WMMA f32 16x16x4 distances to all
// 512 centroids of head h, argmin + errs2 + vecs_hat gather + l_commit.
// Block = 256 threads (8 waves) = 128 rows; 32 centroid groups staged in LDS.
// ---------------------------------------------------------------------------
__global__ __launch_bounds__(256) void vq_main(const float* __restrict__ vecs,
                                               const float* __restrict__ loss_mask,
                                               const float* __restrict__ ws_c,
                                               const float* __restrict__ ws_c2,
                                               float* __restrict__ out_vhat,
                                               float* __restrict__ out_z,
                                               float* __restrict__ out_errs,
                                               float* __restrict__ out_lc) {
    __shared__ float sA[16 * 68];   // 16 centroids x 64, stride 68 (bank-pad)
    __shared__ float sC2[16];
    __shared__ float sRed[8];

    const int tid  = threadIdx.x;
    const int wid  = tid >> 5;
    const int lane = tid & 31;
    const int n    = lane & 15;     // vector column (B/N) == centroid row (A/M)
    const int half = lane >> 4;

    const int bIdx  = blockIdx.x;
    const int bh    = bIdx >> 5;              // 32 blocks per (b,h)
    const int tileL = (bIdx & 31) * 128;
    const int b     = bh >> 3;
    const int h     = bh & 7;

    const int    l     = tileL + wid * 16 + n;
    const float* vrow  = vecs + ((size_t)bh * L_ + l) * DK_;
    const float* chead = ws_c + (size_t)h * S_ * DK_;
    const float* c2head = ws_c2 + h * S_;

    // Load B fragment (vector l, K=64 as 16 steps of K=4) + |v|^2.
    // Lane layout (32-bit B 4x16): VGPR v holds K = 4t + 2*half + v.
    v2f bf[16];
    float v2p = 0.0f;
    #pragma unroll
    for (int t = 0; t < 16; ++t) {
        bf[t] = *(const v2f*)(vrow + 4 * t + 2 * half);
        v2p += bf[t].x * bf[t].x + bf[t].y * bf[t].y;
    }
    const float v2 = v2p + __shfl_xor(v2p, 16, 32);

    float best = 3.4e38f;
    int   bidx = 0;

    for (int sg = 0; sg < 32; ++sg) {
        __syncthreads();
        {   // Stage 16 centroids (1024 floats) into padded LDS: one v4f/thread.
            const int e = tid * 4;
            const int m = e >> 6;
            const int k = e & 63;
            v4f val = *(const v4f*)(chead + (size_t)(sg * 16 + m) * DK_ + k);
            *(v4f*)(&sA[m * 68 + k]) = val;
            if (tid < 16) sC2[tid] = c2head[sg * 16 + tid];
        }
        __syncthreads();

        v8f acc = {0.f, 0.f, 0.f, 0.f, 0.f, 0.f, 0.f, 0.f};
        #pragma unroll
        for (int t = 0; t < 16; ++t) {
            v2f af = *(const v2f*)(&sA[n * 68 + 4 * t + 2 * half]);
            acc = __builtin_amdgcn_wmma_f32_16x16x4_f32(
                false, af, false, bf[t], (short)0, acc, false, false);
        }

        // D layout: lane(half) vgpr r holds centroid m = r + 8*half, col n.
        #pragma unroll
        for (int r = 0; r < 8; ++r) {
            const int   mloc  = r + 8 * half;
            const float score = sC2[mloc] - 2.0f * acc[r];
            const int   gi    = sg * 16 + mloc;
            if (score < best) { best = score; bidx = gi; }
        }
    }

    // Merge the two half-lane partial argmins (tie -> smaller index, matching
    // jnp.argmin first-occurrence).
    {
        const float ob = __shfl_xor(best, 16, 32);
        const int   oi = __shfl_xor(bidx, 16, 32);
        if (ob < best || (ob == best && oi < bidx)) { best = ob; bidx = oi; }
    }
    const float err = fmaxf(v2 + best, 0.0f);

    const size_t rowbase = (size_t)bh * L_ + l;
    if (half == 0) {
        out_z[rowbase]    = (float)bidx;
        out_errs[rowbase] = err;
    }

    // l_commit = sum(mask * errs2) / (B*L): wave reduce -> block reduce -> atomic.
    float lm = (half == 0) ? loss_mask[(size_t)b * L_ + l] * err : 0.0f;
    #pragma unroll
    for (int off = 16; off > 0; off >>= 1) lm += __shfl_xor(lm, off, 32);
    if (lane == 0) sRed[wid] = lm;
    __syncthreads();
    if (tid == 0) {
        float s = 0.0f;
        #pragma unroll
        for (int w = 0; w < 8; ++w) s += sRed[w];
        atomicAdd(out_lc, s * (1.0f / (float)(B_ * L_)));
    }

    // vecs_hat = c[h][z]: lane-cooperative coalesced row copy (L2-hot source).
    #pragma unroll 1
    for (int r = 0; r < 16; ++r) {
        const int zi = __shfl(bidx, r, 32);
        const int lr = tileL + wid * 16 + r;
        const float* src = chead + (size_t)zi * DK_ + 2 * lane;
        float*       dst = out_vhat + ((size_t)bh * L_ + lr) * DK_ + 2 * lane;
        *(v2f*)dst = *(const v2f*)src;
    }
}

// ---------------------------------------------------------------------------
extern "C" void kernel_launch(void* const* d_in, const int* in_sizes, int n_in,
                              void* d_out, int out_size, void* d_ws, size_t ws_size,
                              hipStream_t stream) {
    const float* vecs      = (const float*)d_in[0];  // (B,H,L,DK) f32
    const float* c_sum     = (const float*)d_in[1];  // (H,S,DK)   f32
    const float* c_count   = (const float*)d_in[2];  // (H,S)      f32
    const float* loss_mask = (const float*)d_in[3];  // (B,L)      f32
    // d_in[4]=n_device, d_in[5]=n_block_per_update: only feed l_codebook,
    // which is numerically zero -> ignored.

    float* out = (float*)d_out;
    float* wsf = (float*)d_ws;
    float* ws_c  = wsf;                                  // H*S*DK floats (1 MB)
    float* ws_c2 = wsf + (size_t)H_ * S_ * DK_;          // H*S floats

    const size_t N1 = (size_t)B_ * H_ * L_ * DK_;        // vecs_hat
    const size_t N2 = (size_t)B_ * H_ * L_;              // z / errs2
    float* out_vhat = out;
    float* out_z    = out + N1;
    float* out_lc   = out + N1 + N2;
    float* out_lcb  = out_lc + 1;
    float* out_errs = out_lc + 2;

    vq_init_scalars<<<1, 1, 0, stream>>>(out_lc, out_lcb);
    vq_prep<<<(H_ * S_) / 8, 256, 0, stream>>>(c_sum, c_count, ws_c, ws_c2);
    vq_main<<<B_ * H_ * (L_ / 128), 256, 0, stream>>>(
        vecs, loss_mask, ws_c, ws_c2, out_vhat, out_z, out_errs, out_lc);
}